// IAutoRec_43130061586863
// MI455X (gfx1250) — compile-verified
//
#include <hip/hip_runtime.h>
#include <hip/hip_bf16.h>

typedef __attribute__((ext_vector_type(2))) float v2f;
typedef __attribute__((ext_vector_type(8))) float v8f;

#define D_DIM 64
#define N_ITEMS_C 20000

// ---------------------------------------------------------------------------
// Phase 1: scatter SpMM.  h_pre[iid] += r * v[uid]  (rows of 64 f32)
// One wave handles 32 edges: coalesced index loads, __shfl broadcast, each
// lane owns a contiguous float2 slice of the 256B row -> coalesced gather of
// v and contiguous global_atomic_add_f32 into L2-resident h_pre.
// ---------------------------------------------------------------------------
__global__ void iautorec_spmm_scatter(const int* __restrict__ uid_in,
                                      const int* __restrict__ iid_in,
                                      const float* __restrict__ r_in,
                                      const float* __restrict__ v,
                                      float* __restrict__ h_pre,
                                      int nnz) {
    const int lane   = threadIdx.x & 31;
    const int wave   = (blockIdx.x * blockDim.x + threadIdx.x) >> 5;
    const int nwaves = (gridDim.x * blockDim.x) >> 5;

    for (int base = wave * 32; base < nnz; base += nwaves * 32) {
        const int e = base + lane;
        int   u = 0, it = 0;
        float rv = 0.0f;
        if (e < nnz) { u = uid_in[e]; it = iid_in[e]; rv = r_in[e]; }
        const int cnt = (nnz - base) < 32 ? (nnz - base) : 32;
        for (int j = 0; j < cnt; ++j) {
            const int   uj = __shfl(u,  j, 32);
            const int   ij = __shfl(it, j, 32);
            const float rj = __shfl(rv, j, 32);
            const float2 vv = *reinterpret_cast<const float2*>(
                v + (size_t)uj * D_DIM + lane * 2);
            float* dst = h_pre + (size_t)ij * D_DIM + lane * 2;
            atomicAdd(dst + 0, rj * vv.x);
            atomicAdd(dst + 1, rj * vv.y);
        }
    }
}

// ---------------------------------------------------------------------------
// Phase 2: h = sigmoid(h_pre + mu), in place. 1.28M elements.
// ---------------------------------------------------------------------------
__global__ void iautorec_sigmoid(float* __restrict__ h,
                                 const float* __restrict__ mu,
                                 int n) {
    const int i = blockIdx.x * blockDim.x + threadIdx.x;
    if (i < n) {
        const float x = h[i] + mu[i & (D_DIM - 1)];
        h[i] = 1.0f / (1.0f + __expf(-x));
    }
}

// ---------------------------------------------------------------------------
// Phase 3: out[p] = dot(h[iid_out[p]], w[uid_out[p]]) + b[uid_out[p]]
// 16 pairs per wave via V_WMMA_F32_16X16X4_F32:
//   A (16x4)  = h rows of the 16 pairs for one K-chunk
//       lane L: M = L&15, holds K = {0,1} (lanes 0-15) or {2,3} (lanes 16-31)
//   B (4x16)  = w rows of the 16 pairs (pair index = N), mirrored K split
//   C (16x16) accumulates over 16 chunks (K = 64 total); we need only diag.
// Diagonal layout of f32 C/D: pair p(0..7) -> (VGPR p, lane p);
//                             pair p+8    -> (VGPR p, lane p+24).
// Wasted off-diagonal FLOPs are free: the kernel is L2-bandwidth bound.
// ---------------------------------------------------------------------------
__global__ void iautorec_pair_dot_wmma(const int* __restrict__ uid_out,
                                       const int* __restrict__ iid_out,
                                       const float* __restrict__ h,
                                       const float* __restrict__ w,
                                       const float* __restrict__ b,
                                       float* __restrict__ out,
                                       int p_total) {
    const int lane    = threadIdx.x & 31;
    const int wave    = (blockIdx.x * blockDim.x + threadIdx.x) >> 5;
    const int nwaves  = (gridDim.x * blockDim.x) >> 5;
    const int ngroups = p_total >> 4;           // 16 pairs per group
    const int m       = lane & 15;              // pair within group this lane serves
    const int khalf   = (lane >> 4) << 1;       // K offset 0 or 2 within a chunk

    for (int g = wave; g < ngroups; g += nwaves) {   // wave-uniform: EXEC stays all-1s
        const int gbase = g << 4;
        const int it = iid_out[gbase + m];
        const int u  = uid_out[gbase + m];
        const float* hrow = h + (size_t)it * D_DIM + khalf;
        const float* wrow = w + (size_t)u  * D_DIM + khalf;

        v8f c = {};
#pragma unroll
        for (int kc = 0; kc < 16; ++kc) {       // 16 chunks x K=4 = 64
            const v2f a  = *reinterpret_cast<const v2f*>(hrow + kc * 4);
            const v2f bb = *reinterpret_cast<const v2f*>(wrow + kc * 4);
            c = __builtin_amdgcn_wmma_f32_16x16x4_f32(
                    /*neg_a=*/false, a, /*neg_b=*/false, bb,
                    /*c_mod=*/(short)0, c, /*reuse_a=*/false, /*reuse_b=*/false);
        }

        // Extract diagonal + bias; predicated stores after the WMMA chain.
#pragma unroll
        for (int i = 0; i < 8; ++i) {
            if (lane == i)
                out[gbase + i]     = c[i] + b[uid_out[gbase + i]];
            if (lane == i + 24)
                out[gbase + i + 8] = c[i] + b[uid_out[gbase + i + 8]];
        }
    }
}

extern "C" void kernel_launch(void* const* d_in, const int* in_sizes, int n_in,
                              void* d_out, int out_size, void* d_ws, size_t ws_size,
                              hipStream_t stream) {
    const int*   uid_in  = (const int*)d_in[0];
    const int*   iid_in  = (const int*)d_in[1];
    const float* r_in    = (const float*)d_in[2];
    const int*   uid_out = (const int*)d_in[3];
    const int*   iid_out = (const int*)d_in[4];
    const float* v       = (const float*)d_in[5];
    const float* mu      = (const float*)d_in[6];
    const float* w       = (const float*)d_in[7];
    const float* b       = (const float*)d_in[8];
    float*       out     = (float*)d_out;

    const int nnz = in_sizes[0];
    const int P   = in_sizes[3];

    float* h = (float*)d_ws;                    // N_ITEMS x 64 f32 = 5.1 MB
    const size_t hbytes = (size_t)N_ITEMS_C * D_DIM * sizeof(float);

    // Deterministic per launch: re-zero the accumulator every call.
    hipMemsetAsync(h, 0, hbytes, stream);

    {   // Phase 1: one wave-iteration per 32 edges, 8 waves / 256-thread block.
        const int groups = (nnz + 31) / 32;
        const int blocks = (groups + 7) / 8;
        iautorec_spmm_scatter<<<blocks, 256, 0, stream>>>(uid_in, iid_in, r_in,
                                                          v, h, nnz);
    }
    {   // Phase 2: elementwise sigmoid.
        const int n = N_ITEMS_C * D_DIM;
        iautorec_sigmoid<<<(n + 255) / 256, 256, 0, stream>>>(h, mu, n);
    }
    {   // Phase 3: 16 pairs per wave, WMMA f32 16x16x4 accumulation.
        const int ngroups = P / 16;
        const int blocks  = (ngroups + 7) / 8;
        iautorec_pair_dot_wmma<<<blocks, 256, 0, stream>>>(uid_out, iid_out,
                                                           h, w, b, out, P);
    }
}